// CSOSController_9921374454215
// MI455X (gfx1250) — compile-verified
//
#include <hip/hip_runtime.h>
#include <hip/hip_bf16.h>

// B=64, D_IN=256, N=512, D_OUT=64, DECAY=0.1
// Dominant cost: einsum('ijk,bi,bj->bk') == GEMM [64 x 262144] x [262144 x 512]
// with A[b, i*512+j] = m[b,i]*m[b,j] built on the fly from m staged in LDS and
// the B-matrix being W viewed row-major as [N^2, N]. W = 512MB f32 -> ~22us at
// 23.3 TB/s; ~17.2 GFLOP of f32 WMMA 16x16x4. Bandwidth-bound by design.
// Each wave owns TWO adjacent 16x16 C tiles so it consumes full 128B lines of W
// itself (no reliance on cross-WG L2 timing) and reuses each A fragment twice.

typedef float v2f __attribute__((ext_vector_type(2)));
typedef float v8f __attribute__((ext_vector_type(8)));

#define BATCH   64
#define DIN     256
#define NDIM    512
#define DOUT    64
#define NSPLIT  32              // split-K over outer index i: 32 splits x 16 i's
#define MPITCH  516             // LDS row pitch for m: (4*row + j) mod 64 -> conflict-free

// ---------------- Kernel 1: encoder + decay mix -> m ----------------
// m[b,n] = 0.9*state[b,n] + relu(x[b,:] . enc_w[n,:] + enc_b[n])
__global__ __launch_bounds__(256) void cdna5_encode(
    const float* __restrict__ x, const float* __restrict__ state,
    const float* __restrict__ enc_w, const float* __restrict__ enc_b,
    float* __restrict__ m)
{
    const int idx = blockIdx.x * 256 + threadIdx.x;     // 0 .. 32767
    const int b = idx >> 9;
    const int n = idx & (NDIM - 1);
    const float4* xr = (const float4*)(x + b * DIN);
    const float4* wr = (const float4*)(enc_w + n * DIN);
    float acc = enc_b[n];
    #pragma unroll 8
    for (int k = 0; k < DIN / 4; ++k) {
        float4 xv = xr[k];
        float4 wv = wr[k];
        acc = fmaf(xv.x, wv.x, acc);
        acc = fmaf(xv.y, wv.y, acc);
        acc = fmaf(xv.z, wv.z, acc);
        acc = fmaf(xv.w, wv.w, acc);
    }
    m[idx] = 0.9f * state[idx] + fmaxf(acc, 0.0f);
}

// ---------------- Kernel 2: cubic tensor contraction via WMMA ----------------
// grid = 8 ktiles * NSPLIT blocks of 256 threads (8 waves).
// Block (kt, split): waves 0-3 cover k columns kt*64..+31 (rows 16w..16w+15),
// waves 4-7 cover kt*64+32..+63. Each wave holds two 16x16 f32 accumulators.
// part[split][0..63][kt*64 .. kt*64+63] = sum over i in split, all j of
//   (m[b,j]*m[b,i]) * W[i,j,k].
__global__ __launch_bounds__(256) void cdna5_contract(
    const float* __restrict__ W, const float* __restrict__ m,
    float* __restrict__ part)
{
    extern __shared__ float mlds[];                 // [64][MPITCH] ~132KB
    const int kt    = blockIdx.x & 7;               // 8 ktiles of 64 columns
    const int split = blockIdx.x >> 3;              // 32 i-splits
    const int tid   = threadIdx.x;

    // Stage m (64x512) into padded LDS with 128-bit stores.
    {
        const float4* mg = (const float4*)m;
        for (int q = tid; q < (BATCH * NDIM) / 4; q += 256) {
            const int r  = q >> 7;                  // 128 float4 per row
            const int jc = (q & 127) * 4;
            *(float4*)&mlds[r * MPITCH + jc] = mg[q];
        }
    }
    __syncthreads();

    const int wave = tid >> 5;
    const int lane = tid & 31;
    const int half = lane >> 4;                     // K-half per ISA A/B layout
    const int ln   = lane & 15;
    const int wv   = wave & 3;                      // M-tile index
    const int hi   = wave >> 2;                     // which 32-column group
    const int k0   = kt * 64 + hi * 32;             // first of 32 columns
    const int row  = wv * 16 + ln;                  // C row this lane serves
    const float* __restrict__ mrow = &mlds[row * MPITCH];
    const int hoff  = half * 2;                     // j offset for this half
    const int whoff = half * 2 * NDIM;              // W offset (2 rows of j)

    v8f c0 = {0.f, 0.f, 0.f, 0.f, 0.f, 0.f, 0.f, 0.f};
    v8f c1 = {0.f, 0.f, 0.f, 0.f, 0.f, 0.f, 0.f, 0.f};

    const int i0 = split * (NDIM / NSPLIT);
    for (int ii = 0; ii < NDIM / NSPLIT; ++ii) {
        const int i = i0 + ii;
        const float si = mrow[i];                   // m[row, i] (LDS broadcast)
        const float* __restrict__ wp =
            W + (size_t)i * NDIM * NDIM + k0 + ln + whoff;
        // Prefetch the next i-slab's first rows for this k-group.
        __builtin_prefetch(wp + (size_t)NDIM * NDIM, 0, 1);
        #pragma unroll 4
        for (int j0 = 0; j0 < NDIM; j0 += 4) {
            v2f a, b0, b1;
            // ds_load_2addr_b32: two consecutive m values; scale by m[row,i]
            a.x = mrow[j0 + hoff]     * si;         // A[M=row][K = half*2]
            a.y = mrow[j0 + hoff + 1] * si;         // A[M=row][K = half*2+1]
            // four b32 global loads: W[i, j0+half*2 + {0,1}, k0 + {0,16} + ln]
            // lanes 0-15 + both tiles => full contiguous 128B per j-row.
            b0.x = wp[0];
            b0.y = wp[NDIM];
            b1.x = wp[16];
            b1.y = wp[NDIM + 16];
            c0 = __builtin_amdgcn_wmma_f32_16x16x4_f32(
                     false, a, false, b0, (short)0, c0, false, false);
            c1 = __builtin_amdgcn_wmma_f32_16x16x4_f32(
                     false, a, false, b1, (short)0, c1, false, false);
            wp += 4 * NDIM;
        }
    }

    // Store partial tiles: VGPR v holds (M = 16*wv + v + 8*half, N = ln).
    float* __restrict__ out = part + (size_t)split * BATCH * NDIM;
    #pragma unroll
    for (int v = 0; v < 8; ++v) {
        const int r = wv * 16 + v + 8 * half;
        out[r * NDIM + k0 + ln]      = c0[v];
        out[r * NDIM + k0 + 16 + ln] = c1[v];
    }
}

// ---------------- Kernel 3: reduce splits, relu, L1-normalize, decode, argmax --
__global__ __launch_bounds__(256) void cdna5_finalize(
    const float* __restrict__ m, const float* __restrict__ part,
    const float* __restrict__ dec_w, const float* __restrict__ dec_b,
    float* __restrict__ logits, float* __restrict__ new_state,
    int* __restrict__ action)
{
    __shared__ float ns[NDIM];
    __shared__ float red[256];
    __shared__ float lg[DOUT];
    const int b = blockIdx.x;
    const int tid = threadIdx.x;

    float lsum = 0.f;
    for (int j = tid; j < NDIM; j += 256) {
        float acc = 0.f;
        #pragma unroll
        for (int s = 0; s < NSPLIT; ++s)            // fixed-order split-K reduce
            acc += part[((size_t)s * BATCH + b) * NDIM + j];
        float v = fmaxf(m[b * NDIM + j] + 0.5f * acc, 0.f);
        ns[j] = v;
        lsum += v;
    }
    red[tid] = lsum;
    __syncthreads();
    for (int off = 128; off > 0; off >>= 1) {
        if (tid < off) red[tid] += red[tid + off];
        __syncthreads();
    }
    const float denom = red[0] + 1e-8f;
    for (int j = tid; j < NDIM; j += 256) {
        float v = ns[j] / denom;                    // division, as in reference
        ns[j] = v;
        new_state[b * NDIM + j] = v;
    }
    __syncthreads();
    if (tid < DOUT) {
        const float4* dw = (const float4*)(dec_w + tid * NDIM);
        float acc = dec_b[tid];
        #pragma unroll 8
        for (int j = 0; j < NDIM / 4; ++j) {
            float4 w4 = dw[j];
            acc = fmaf(ns[4 * j + 0], w4.x, acc);
            acc = fmaf(ns[4 * j + 1], w4.y, acc);
            acc = fmaf(ns[4 * j + 2], w4.z, acc);
            acc = fmaf(ns[4 * j + 3], w4.w, acc);
        }
        lg[tid] = acc;
        logits[b * DOUT + tid] = acc;
    }
    __syncthreads();
    if (tid == 0) {                                 // argmax: first max wins
        int best = 0;
        float bv = lg[0];
        for (int o = 1; o < DOUT; ++o)
            if (lg[o] > bv) { bv = lg[o]; best = o; }
        action[b] = best;
    }
}

extern "C" void kernel_launch(void* const* d_in, const int* in_sizes, int n_in,
                              void* d_out, int out_size, void* d_ws, size_t ws_size,
                              hipStream_t stream) {
    (void)in_sizes; (void)n_in; (void)out_size; (void)ws_size;
    const float* x      = (const float*)d_in[0];
    const float* state  = (const float*)d_in[1];
    const float* enc_w  = (const float*)d_in[2];
    const float* enc_b  = (const float*)d_in[3];
    const float* W      = (const float*)d_in[4];
    const float* dec_w  = (const float*)d_in[5];
    const float* dec_b  = (const float*)d_in[6];

    float* out       = (float*)d_out;
    float* logits    = out;                              // 64*64
    float* new_state = out + BATCH * DOUT;               // 64*512
    int*   action    = (int*)(out + BATCH * DOUT + BATCH * NDIM);  // 64 int32

    float* m    = (float*)d_ws;                          // 64*512 f32    (128 KB)
    float* part = m + BATCH * NDIM;                      // 32*64*512 f32 (4 MB)

    cdna5_encode<<<(BATCH * NDIM) / 256, 256, 0, stream>>>(x, state, enc_w, enc_b, m);

    const size_t mlds_bytes = (size_t)BATCH * MPITCH * sizeof(float);  // ~132 KB
    cdna5_contract<<<8 * NSPLIT, 256, mlds_bytes, stream>>>(W, m, part);

    cdna5_finalize<<<BATCH, 256, 0, stream>>>(m, part, dec_w, dec_b,
                                              logits, new_state, action);
}